// HiveMind_GCN_v2_62534723830374
// MI455X (gfx1250) — compile-verified
//
#include <hip/hip_runtime.h>

#define NN      150000
#define EE      2400000
#define GG      128
#define HH      128
#define EMB     512
#define BN_EPS  1e-5f

typedef __attribute__((ext_vector_type(2))) float v2f;
typedef __attribute__((ext_vector_type(8))) float v8f;

// ---------------------------------------------------------------- utilities
__global__ void k_fill_f32(float* __restrict__ p, float v, int n) {
    int i = blockIdx.x * 256 + threadIdx.x;
    if (i < n) p[i] = v;
}

__global__ void k_rsqrt_inplace(float* __restrict__ p, int n) {
    int i = blockIdx.x * 256 + threadIdx.x;
    if (i < n) {
        float d = p[i];
        p[i] = (d > 0.0f) ? rsqrtf(d) : 0.0f;
    }
}

// deg[dst] += 1 for each edge (deg buffer pre-filled with 1.0 for self-loops)
__global__ void k_deg_edges(const int* __restrict__ ei, float* __restrict__ deg, int e_cnt) {
    int e = blockIdx.x * 256 + threadIdx.x;
    if (e < e_cnt) atomicAdd(&deg[ei[e_cnt + e]], 1.0f);   // dst row of edge_index
}

// ---------------------------------------------------------------- layer 1: x(N,3) @ W(3,128)
__global__ void k_feat_transform(const float* __restrict__ x, const float* __restrict__ W,
                                 float* __restrict__ h, int total) {
    int idx = blockIdx.x * 256 + threadIdx.x;
    if (idx >= total) return;
    int n = idx >> 7, j = idx & 127;
    float acc = x[n * 3 + 0] * W[0 * HH + j]
              + x[n * 3 + 1] * W[1 * HH + j]
              + x[n * 3 + 2] * W[2 * HH + j];
    h[idx] = acc;
}

// ---------------------------------------------------------------- B pre-pack
// Bpack[((nt*32 + kq)*32 + lane)*2 + j] = B[(4*kq + 2*(lane>>4) + j) * Nc + nt*16 + (lane&15)]
// -> per (wave,nt): lane's fragment for k-step kq is one float2 at base + lane*8 + kq*256 bytes.
__global__ void k_pack_B(const float* __restrict__ B, float* __restrict__ Bpack, int Nc) {
    int idx = blockIdx.x * 256 + threadIdx.x;
    int total = (Nc >> 4) << 11;                 // ntiles * 2048
    if (idx >= total) return;
    int j    = idx & 1;
    int lane = (idx >> 1) & 31;
    int kq   = (idx >> 6) & 31;
    int nt   = idx >> 11;
    int k = 4 * kq + 2 * (lane >> 4) + j;
    int n = (nt << 4) + (lane & 15);
    Bpack[idx] = B[(size_t)k * Nc + n];
}

// ---------------------------------------------------------------- WMMA fp32 GEMM
// C(M,Nc) = A(M,128) @ B(128,Nc) [+bias][, relu].  Block = 256 thr = 8 waves.
// A tile staged in LDS (row pad 132 -> 8B aligned, banks spread); B pre-packed so
// every fragment is one global_load_b64 at an immediate offset.
__global__ void k_gemm_wmma_f32(const float* __restrict__ A, const float* __restrict__ Bpack,
                                const float* __restrict__ bias, float* __restrict__ C,
                                int M, int Nc, int relu) {
    __shared__ float As[16][132];
    const int K = 128;
    int tid = threadIdx.x;
    int m0  = blockIdx.x * 16;

    for (int idx = tid; idx < 16 * K; idx += 256) {  // coalesced A-tile stage
        int r = idx >> 7, c = idx & 127;
        int gr = m0 + r;
        As[r][c] = (gr < M) ? A[(size_t)gr * K + c] : 0.0f;
    }
    __syncthreads();

    int wave = tid >> 5;
    int lane = tid & 31;
    int row  = lane & 15;                 // A row within tile
    int koff = (lane >> 4) * 2;           // K sub-offset per half-wave
    int col  = lane & 15;                 // B/C column within tile
    int ntiles = Nc >> 4;

    const float2* Arow = (const float2*)(&As[row][koff]);   // k-step stride = 2 float2

    for (int nt = wave; nt < ntiles; nt += 8) {
        const float2* Bp = (const float2*)(Bpack + ((size_t)nt << 11)) + lane;
        v8f acc = {};
#pragma unroll
        for (int kq = 0; kq < 32; ++kq) {
            float2 af = Arow[kq * 2];     // ds_load_b64, imm offset kq*16
            float2 bf = Bp[kq * 32];      // global_load_b64, imm offset kq*256
            v2f a; a.x = af.x; a.y = af.y;
            v2f b; b.x = bf.x; b.y = bf.y;
            acc = __builtin_amdgcn_wmma_f32_16x16x4_f32(
                      false, a, false, b, (short)0, acc, false, false);
        }
        int orow0 = m0 + (lane >> 4) * 8;            // C layout: acc[r] -> row r (+8 hi half)
        int gc = (nt << 4) + col;
        float bj = bias ? bias[gc] : 0.0f;
#pragma unroll
        for (int r = 0; r < 8; ++r) {
            int gr = orow0 + r;
            if (gr < M) {
                float v = acc[r] + bj;
                if (relu) v = fmaxf(v, 0.0f);
                C[(size_t)gr * Nc + gc] = v;
            }
        }
    }
}

// ---------------------------------------------------------------- aggregation
// agg[n] = h[n] * dinv[n]^2   (self-loop term, also initializes agg buffer)
__global__ void k_selfloop_init(const float* __restrict__ h, const float* __restrict__ dinv,
                                float* __restrict__ agg, int n_cnt) {
    int t = blockIdx.x * 256 + threadIdx.x;
    int n = t >> 5;
    if (n >= n_cnt) return;
    int lane = t & 31;
    float w = dinv[n]; w *= w;
    float4 v = *((const float4*)(h + (size_t)n * HH) + lane);
    float4 o = { v.x * w, v.y * w, v.z * w, v.w * w };
    *((float4*)(agg + (size_t)n * HH) + lane) = o;
}

// one wave per edge: agg[dst] += h[src] * dinv[src]*dinv[dst]
__global__ void k_edge_aggregate(const int* __restrict__ ei, const float* __restrict__ dinv,
                                 const float* __restrict__ h, float* __restrict__ agg, int e_cnt) {
    long long t = (long long)blockIdx.x * 256 + threadIdx.x;
    int e = (int)(t >> 5);
    if (e >= e_cnt) return;
    int lane = (int)(t & 31);
    int s = ei[e];
    int d = ei[e_cnt + e];
    float nrm = dinv[s] * dinv[d];
    float4 v = *((const float4*)(h + (size_t)s * HH) + lane);
    float* ap = agg + (size_t)d * HH + lane * 4;
    atomicAdd(ap + 0, v.x * nrm);
    atomicAdd(ap + 1, v.y * nrm);
    atomicAdd(ap + 2, v.z * nrm);
    atomicAdd(ap + 3, v.w * nrm);
}

// ---------------------------------------------------------------- batch norm
__global__ void k_bn_stats(const float* __restrict__ y, float* __restrict__ stats, int rows) {
    __shared__ float ls[256], lq[256];
    int tid = threadIdx.x;
    int col = tid & 127, half = tid >> 7;
    float s = 0.0f, q = 0.0f;
    for (int r = blockIdx.x * 2 + half; r < rows; r += gridDim.x * 2) {
        float v = y[(size_t)r * HH + col];
        s += v; q += v * v;
    }
    ls[tid] = s; lq[tid] = q;
    __syncthreads();
    if (tid < 128) {
        atomicAdd(&stats[col],       ls[tid] + ls[tid + 128]);
        atomicAdd(&stats[128 + col], lq[tid] + lq[tid + 128]);
    }
}

__global__ void k_bn_apply_relu(float* __restrict__ y, const float* __restrict__ stats,
                                const float* __restrict__ g, const float* __restrict__ be,
                                int total, float inv_rows) {
    int idx = blockIdx.x * 256 + threadIdx.x;
    if (idx >= total) return;
    int col = idx & 127;
    float mu  = stats[col] * inv_rows;
    float var = stats[128 + col] * inv_rows - mu * mu;
    float v = (y[idx] - mu) * rsqrtf(var + BN_EPS) * g[col] + be[col];
    y[idx] = fmaxf(v, 0.0f);
}

// ---------------------------------------------------------------- global mean pool
__global__ void k_pool_accum(const float* __restrict__ y, const int* __restrict__ batch,
                             float* __restrict__ psum, float* __restrict__ pcnt, int n_cnt) {
    int t = blockIdx.x * 256 + threadIdx.x;
    int n = t >> 5;
    if (n >= n_cnt) return;
    int lane = t & 31;
    int b = batch[n];
    float4 v = *((const float4*)(y + (size_t)n * HH) + lane);
    float* p = psum + (size_t)b * HH + lane * 4;
    atomicAdd(p + 0, v.x);
    atomicAdd(p + 1, v.y);
    atomicAdd(p + 2, v.z);
    atomicAdd(p + 3, v.w);
    if (lane == 0) atomicAdd(&pcnt[b], 1.0f);
}

// pooled = psum / max(cnt,1) + b3  (conv3 bias folded here: mean(x+b) = mean(x)+b)
__global__ void k_pool_final(const float* __restrict__ psum, const float* __restrict__ pcnt,
                             const float* __restrict__ b3, float* __restrict__ pooled, int total) {
    int idx = blockIdx.x * 256 + threadIdx.x;
    if (idx >= total) return;
    int gIdx = idx >> 7, col = idx & 127;
    pooled[idx] = psum[idx] / fmaxf(pcnt[gIdx], 1.0f) + b3[col];
}

// ---------------------------------------------------------------- row-wise L2 normalize
__global__ void k_l2norm(const float* __restrict__ p, float* __restrict__ out, int cols) {
    __shared__ float red[256];
    int g = blockIdx.x, tid = threadIdx.x;
    float s = 0.0f;
    for (int c = tid; c < cols; c += 256) {
        float v = p[(size_t)g * cols + c];
        s += v * v;
    }
    red[tid] = s;
    __syncthreads();
    for (int off = 128; off > 0; off >>= 1) {
        if (tid < off) red[tid] += red[tid + off];
        __syncthreads();
    }
    float nrm = fmaxf(sqrtf(red[0]), 1e-12f);
    for (int c = tid; c < cols; c += 256)
        out[(size_t)g * cols + c] = p[(size_t)g * cols + c] / nrm;
}

// ================================================================ host
static inline int cdiv(long long a, int b) { return (int)((a + b - 1) / b); }

extern "C" void kernel_launch(void* const* d_in, const int* in_sizes, int n_in,
                              void* d_out, int out_size, void* d_ws, size_t ws_size,
                              hipStream_t stream) {
    const float* x[2]     = { (const float*)d_in[0], (const float*)d_in[3] };
    const int*   ei[2]    = { (const int*)  d_in[1], (const int*)  d_in[4] };
    const int*   batch[2] = { (const int*)  d_in[2], (const int*)  d_in[5] };
    const float* W1  = (const float*)d_in[6];
    const float* W2  = (const float*)d_in[8];
    const float* W3  = (const float*)d_in[10];
    const float* b3  = (const float*)d_in[11];
    const float* g1  = (const float*)d_in[12];
    const float* be1 = (const float*)d_in[13];
    const float* g2  = (const float*)d_in[14];
    const float* be2 = (const float*)d_in[15];
    const float* Wp1 = (const float*)d_in[16];
    const float* bp1 = (const float*)d_in[17];
    const float* Wp2 = (const float*)d_in[18];
    const float* bp2 = (const float*)d_in[19];
    // b1,b2 (d_in[7], d_in[9]) cancel under training-mode BN and are skipped.

    char* w = (char*)d_ws;
    auto carve = [&](size_t bytes) -> float* {
        float* p = (float*)w;
        w += (bytes + 255) & ~(size_t)255;
        return p;
    };
    float* buf0   = carve((size_t)NN * HH * 4);   // dense-transform output
    float* buf1   = carve((size_t)NN * HH * 4);   // aggregation output
    float* dinv   = carve((size_t)NN * 4);
    float* stats  = carve(256 * 4);
    float* psum   = carve((size_t)GG * HH * 4);
    float* pcnt   = carve((size_t)GG * 4);
    float* pooled = carve((size_t)GG * HH * 4);
    float* p1     = carve((size_t)GG * HH * 4);
    float* p2     = carve((size_t)GG * EMB * 4);
    float* W2p    = carve((size_t)HH * HH * 4);   // WMMA-fragment-packed weights
    float* W3p    = carve((size_t)HH * HH * 4);
    float* Wp1p   = carve((size_t)HH * HH * 4);
    float* Wp2p   = carve((size_t)HH * EMB * 4);

    const int gN   = cdiv(NN, 256);
    const int gNH  = cdiv((long long)NN * HH, 256);
    const int gN32 = cdiv((long long)NN * 32, 256);
    const int gE   = cdiv(EE, 256);
    const int gE32 = cdiv((long long)EE * 32, 256);
    const float inv_rows = 1.0f / (float)NN;

    // ---- pack all GEMM weights into per-lane WMMA fragment order (once per call)
    k_pack_B<<<cdiv(HH * HH, 256), 256, 0, stream>>>(W2,  W2p,  HH);
    k_pack_B<<<cdiv(HH * HH, 256), 256, 0, stream>>>(W3,  W3p,  HH);
    k_pack_B<<<cdiv(HH * HH, 256), 256, 0, stream>>>(Wp1, Wp1p, HH);
    k_pack_B<<<cdiv(HH * EMB, 256), 256, 0, stream>>>(Wp2, Wp2p, EMB);

    for (int br = 0; br < 2; ++br) {
        // ---- degree -> dinv (self-loop counted via init to 1.0)
        k_fill_f32<<<gN, 256, 0, stream>>>(dinv, 1.0f, NN);
        k_deg_edges<<<gE, 256, 0, stream>>>(ei[br], dinv, EE);
        k_rsqrt_inplace<<<gN, 256, 0, stream>>>(dinv, NN);

        // ---- layer 1: h = x @ W1 ; agg ; BN+ReLU
        k_feat_transform<<<gNH, 256, 0, stream>>>(x[br], W1, buf0, NN * HH);
        k_selfloop_init<<<gN32, 256, 0, stream>>>(buf0, dinv, buf1, NN);
        k_edge_aggregate<<<gE32, 256, 0, stream>>>(ei[br], dinv, buf0, buf1, EE);
        k_fill_f32<<<1, 256, 0, stream>>>(stats, 0.0f, 256);
        k_bn_stats<<<2048, 256, 0, stream>>>(buf1, stats, NN);
        k_bn_apply_relu<<<gNH, 256, 0, stream>>>(buf1, stats, g1, be1, NN * HH, inv_rows);

        // ---- layer 2: h = y @ W2 (WMMA) ; agg ; BN+ReLU
        k_gemm_wmma_f32<<<cdiv(NN, 16), 256, 0, stream>>>(buf1, W2p, nullptr, buf0, NN, HH, 0);
        k_selfloop_init<<<gN32, 256, 0, stream>>>(buf0, dinv, buf1, NN);
        k_edge_aggregate<<<gE32, 256, 0, stream>>>(ei[br], dinv, buf0, buf1, EE);
        k_fill_f32<<<1, 256, 0, stream>>>(stats, 0.0f, 256);
        k_bn_stats<<<2048, 256, 0, stream>>>(buf1, stats, NN);
        k_bn_apply_relu<<<gNH, 256, 0, stream>>>(buf1, stats, g2, be2, NN * HH, inv_rows);

        // ---- layer 3: h = y @ W3 (WMMA) ; agg (b3 folded into pooling)
        k_gemm_wmma_f32<<<cdiv(NN, 16), 256, 0, stream>>>(buf1, W3p, nullptr, buf0, NN, HH, 0);
        k_selfloop_init<<<gN32, 256, 0, stream>>>(buf0, dinv, buf1, NN);
        k_edge_aggregate<<<gE32, 256, 0, stream>>>(ei[br], dinv, buf0, buf1, EE);

        // ---- global mean pool (+b3)
        k_fill_f32<<<cdiv(GG * HH, 256), 256, 0, stream>>>(psum, 0.0f, GG * HH);
        k_fill_f32<<<1, 256, 0, stream>>>(pcnt, 0.0f, GG);
        k_pool_accum<<<gN32, 256, 0, stream>>>(buf1, batch[br], psum, pcnt, NN);
        k_pool_final<<<cdiv(GG * HH, 256), 256, 0, stream>>>(psum, pcnt, b3, pooled, GG * HH);

        // ---- projection MLP (WMMA, fused bias/ReLU) + L2 normalize
        k_gemm_wmma_f32<<<cdiv(GG, 16), 256, 0, stream>>>(pooled, Wp1p, bp1, p1, GG, HH, 1);
        k_gemm_wmma_f32<<<cdiv(GG, 16), 256, 0, stream>>>(p1, Wp2p, bp2, p2, GG, EMB, 0);
        k_l2norm<<<GG, 256, 0, stream>>>(p2, (float*)d_out + (size_t)br * GG * EMB, EMB);
    }
}